// Model_57458072486001
// MI455X (gfx1250) — compile-verified
//
#include <hip/hip_runtime.h>
#include <math.h>

#define SEQ   512
#define PRED  96
#define CCH   64
#define TOPM  4
#define NW    6
#define NSAMP 5000
#define BATCH 128
#define NROWS (BATCH * CCH)   // 8192
#define KQ    104             // 48 + 32 + 24 concatenated retrieval K
#define NFUSE 112             // 96 (W_x) + 16 (W_query tile, zero padded)

typedef __attribute__((ext_vector_type(2))) float v2f;
typedef __attribute__((ext_vector_type(8))) float v8f;

// ---- workspace layout (float element offsets) ----
// Weight blocks are pair-interleaved: [k/2][n][2] so a lane's (k,k+1) B pair
// is one aligned 8-byte load.
#define WS_WT    0                            // [256][112][2] fused (W_x|W_query)^T
#define WS_WCAT  (WS_WT   + SEQ * NFUSE)      // [52][96][2]   concat retrieval W^T
#define WS_BSEL  (WS_WCAT + KQ * PRED)        // [96]          b_ret0+b_ret1+b_ret2
#define WS_WPT   (WS_BSEL + PRED)             // [96][96][2]   W_pred^T
#define WS_BASE  (WS_WPT  + 2 * PRED * PRED)  // [8192][96]    base_pred rows (b*64+c)
#define WS_Q     (WS_BASE + NROWS * PRED)     // [8192][6]     quantum input angles
#define WS_SEL   (WS_Q    + NROWS * NW)       // [8192][96]    selected retrieval rows
#define WS_MSEL  (WS_SEL  + NROWS * PRED)     // [8192]        int argmax selection
#define PREP_TOTAL (SEQ * NFUSE + KQ * PRED + PRED + 2 * PRED * PRED)

// ---------------------------------------------------------------------------
// Prep: build pair-interleaved k-major weight layouts.
// ---------------------------------------------------------------------------
__global__ __launch_bounds__(256) void prep_kernel(
    const float* __restrict__ Wq, const float* __restrict__ Wx,
    const float* __restrict__ Wr0, const float* __restrict__ br0,
    const float* __restrict__ Wr1, const float* __restrict__ br1,
    const float* __restrict__ Wr2, const float* __restrict__ br2,
    const float* __restrict__ Wp, float* __restrict__ ws) {
  int i = blockIdx.x * blockDim.x + threadIdx.x;
  if (i < SEQ * NFUSE) {                       // fused (W_x | W_query)^T
    int k = i / NFUSE, n = i % NFUSE;
    float v = 0.0f;
    if (n < PRED)            v = Wx[n * SEQ + k];
    else if (n < PRED + NW)  v = Wq[(n - PRED) * SEQ + k];
    ws[WS_WT + ((k >> 1) * NFUSE + n) * 2 + (k & 1)] = v;
    return;
  }
  i -= SEQ * NFUSE;
  if (i < KQ * PRED) {                         // concat retrieval weights^T
    int k = i / PRED, p = i % PRED;
    float v;
    if (k < 48)      v = Wr0[p * 48 + k];
    else if (k < 80) v = Wr1[p * 32 + (k - 48)];
    else             v = Wr2[p * 24 + (k - 80)];
    ws[WS_WCAT + ((k >> 1) * PRED + p) * 2 + (k & 1)] = v;
    return;
  }
  i -= KQ * PRED;
  if (i < PRED) {                              // summed retrieval biases
    ws[WS_BSEL + i] = br0[i] + br1[i] + br2[i];
    return;
  }
  i -= PRED;
  if (i < 2 * PRED * PRED) {                   // W_pred^T
    int k = i / PRED, p = i % PRED;
    ws[WS_WPT + ((k >> 1) * PRED + p) * 2 + (k & 1)] = Wp[p * (2 * PRED) + k];
  }
}

// ---------------------------------------------------------------------------
// GEMM 1: base_pred (+ query angles), K = 512, one wave per 16-row stripe.
// A is loaded once and reused for all 7 N-tiles via 7 accumulators; each B
// pair is a single b64 load.
// ---------------------------------------------------------------------------
__global__ __launch_bounds__(128) void gemm_base_kernel(
    const float* __restrict__ x_enc, const float* __restrict__ bx,
    const float* __restrict__ bq, float* __restrict__ ws) {
  const int lane = threadIdx.x & 31;
  const int wave = threadIdx.x >> 5;
  const int mt   = blockIdx.x * 4 + wave;   // 0..511
  const int l16  = lane & 15;
  const int hw   = lane >> 4;               // half-wave
  const int koff = hw << 1;
  const int row  = mt * 16 + l16;           // (b,c) row for A operand
  const int b    = row >> 6;
  const int c    = row & 63;
  const float* xb   = x_enc + (size_t)b * SEQ * CCH;
  const float  xoff = xb[511 * CCH + c];
  const v2f* wtp = (const v2f*)(ws + WS_WT) + l16;   // pair table, col l16

  v8f acc[7];
#pragma unroll
  for (int t = 0; t < 7; ++t) acc[t] = (v8f){};

#pragma unroll 2
  for (int k0 = 0; k0 < SEQ; k0 += 4) {
    const int k  = k0 + koff;
    const int p2 = (k0 >> 1) + hw;          // pair index for this half-wave
    v2f a;
    a.x = xb[k * CCH + c] - xoff;
    a.y = xb[(k + 1) * CCH + c] - xoff;
#pragma unroll
    for (int t = 0; t < 7; ++t) {
      const v2f bm = wtp[(size_t)p2 * NFUSE + t * 16];
      acc[t] = __builtin_amdgcn_wmma_f32_16x16x4_f32(false, a, false, bm,
                                                     (short)0, acc[t],
                                                     false, false);
    }
  }

  const int rbase = mt * 16 + (hw << 3);
  float* baseo = ws + WS_BASE;
#pragma unroll
  for (int t = 0; t < 6; ++t) {
    const int n = t * 16 + l16;
    const float bias = bx[n];
#pragma unroll
    for (int r = 0; r < 8; ++r)
      baseo[(size_t)(rbase + r) * PRED + n] = acc[t][r] + bias;
  }
  if (l16 < NW) {
    const float bias = bq[l16];
    float* qo = ws + WS_Q;
#pragma unroll
    for (int r = 0; r < 8; ++r)
      qo[(size_t)(rbase + r) * NW + l16] = acc[6][r] + bias;
  }
}

// ---------------------------------------------------------------------------
// Quantum circuit + Gumbel argmax. State (64 reals) per thread in padded LDS.
// ---------------------------------------------------------------------------
__device__ __forceinline__ void apply_ry(float* s, float cw, float sw, int bit) {
  const int stride = 1 << bit;
  for (int p = 0; p < 32; ++p) {
    const int lo = p & (stride - 1);
    const int i0 = ((p >> bit) << (bit + 1)) | lo;
    const int i1 = i0 | stride;
    const float a0 = s[i0], a1 = s[i1];
    s[i0] = cw * a0 - sw * a1;
    s[i1] = sw * a0 + cw * a1;
  }
}

__global__ __launch_bounds__(64) void quantum_kernel(
    const float* __restrict__ qmat, const float* __restrict__ ang,
    const int* __restrict__ wires, const float* __restrict__ Wprob,
    const float* __restrict__ bprob, const float* __restrict__ gum,
    int* __restrict__ msel) {
  __shared__ float st[64 * 65];              // 65-stride padding: no bank conflicts
  const int tid = threadIdx.x;
  const int row = blockIdx.x * 64 + tid;
  float* s = st + tid * 65;

  for (int i = 0; i < 64; ++i) s[i] = 0.0f;
  s[0] = 1.0f;

  // data-encoding RY gates (wire w acts on bit 5-w)
  for (int w = 0; w < NW; ++w) {
    const float th = qmat[(size_t)row * NW + w] * 0.5f;
    apply_ry(s, cosf(th), sinf(th), 5 - w);
  }
  // fixed ansatz RY gates
  for (int k = 0; k < 3 * NW; ++k) {
    const float th = ang[k] * 0.5f;
    apply_ry(s, cosf(th), sinf(th), 5 - wires[k]);
  }

  float e[NW];
  for (int w = 0; w < NW; ++w) e[w] = 0.0f;
  for (int i = 0; i < 64; ++i) {
    const float p = s[i] * s[i];
    for (int w = 0; w < NW; ++w)
      e[w] += ((i >> (5 - w)) & 1) ? -p : p;
  }

  // argmax(softmax((logits+g)/T)) == argmax(logits+g); first-max wins like jnp.argmax
  int best = 0;
  float bv = -INFINITY;
  for (int m = 0; m < TOPM; ++m) {
    float v = bprob[m] + gum[(size_t)row * TOPM + m];
    for (int w = 0; w < NW; ++w) v += e[w] * Wprob[m * NW + w];
    if (v > bv) { bv = v; best = m; }
  }
  msel[row] = best;
}

// ---------------------------------------------------------------------------
// GEMM 2: selected retrieval, K = 104, one wave per 16-row stripe; gathered
// table elements loaded once, reused across all 6 N-tiles.
// ---------------------------------------------------------------------------
__device__ __forceinline__ long tab_base(int k) {
  int i, r;
  if (k < 48)      { i = 0; r = k * 2; }
  else if (k < 80) { i = 1; r = (k - 48) * 3; }
  else             { i = 2; r = (k - 80) * 4; }
  return (long)i * NSAMP * TOPM * PRED * CCH + (long)r * CCH;
}

__global__ __launch_bounds__(128) void gemm_sel_kernel(
    const float* __restrict__ table, const int* __restrict__ index,
    float* __restrict__ ws) {
  const int lane = threadIdx.x & 31;
  const int wave = threadIdx.x >> 5;
  const int mt   = blockIdx.x * 4 + wave;   // 0..511
  const int l16  = lane & 15;
  const int hw   = lane >> 4;
  const int koff = hw << 1;
  const int row  = mt * 16 + l16;
  const int b    = row >> 6;
  const int c    = row & 63;
  const int* msel = (const int*)(ws + WS_MSEL);
  const long idxterm = (long)index[b] * (TOPM * PRED * CCH) +
                       (long)msel[row] * (PRED * CCH) + c;
  const v2f* wcp = (const v2f*)(ws + WS_WCAT) + l16;

  v8f acc[6];
#pragma unroll
  for (int t = 0; t < 6; ++t) acc[t] = (v8f){};

#pragma unroll 2
  for (int k0 = 0; k0 < KQ; k0 += 4) {
    const int k  = k0 + koff;
    const int p2 = (k0 >> 1) + hw;
    v2f a;
    a.x = table[tab_base(k) + idxterm];
    a.y = table[tab_base(k + 1) + idxterm];
#pragma unroll
    for (int t = 0; t < 6; ++t) {
      const v2f bm = wcp[(size_t)p2 * PRED + t * 16];
      acc[t] = __builtin_amdgcn_wmma_f32_16x16x4_f32(false, a, false, bm,
                                                     (short)0, acc[t],
                                                     false, false);
    }
  }

  const int rbase = mt * 16 + (hw << 3);
  float* out = ws + WS_SEL;
#pragma unroll
  for (int t = 0; t < 6; ++t) {
    const int n = t * 16 + l16;
    const float bias = ws[WS_BSEL + n];
#pragma unroll
    for (int r = 0; r < 8; ++r)
      out[(size_t)(rbase + r) * PRED + n] = acc[t][r] + bias;
  }
}

// ---------------------------------------------------------------------------
// GEMM 3: final projection, K = 192 = [base | selected], + b_pred + x_offset.
// A pair is one b64 load (rows are contiguous); B pairs are b64 loads.
// ---------------------------------------------------------------------------
__global__ __launch_bounds__(128) void gemm_final_kernel(
    const float* __restrict__ x_enc, const float* __restrict__ bpred,
    const float* __restrict__ ws, float* __restrict__ out) {
  const int lane = threadIdx.x & 31;
  const int wave = threadIdx.x >> 5;
  const int mt   = blockIdx.x * 4 + wave;
  const int l16  = lane & 15;
  const int hw   = lane >> 4;
  const int koff = hw << 1;
  const int row  = mt * 16 + l16;
  const float* arow = ws + WS_BASE + (size_t)row * PRED;
  const float* srow = ws + WS_SEL  + (size_t)row * PRED;
  const v2f* wpp = (const v2f*)(ws + WS_WPT) + l16;

  v8f acc[6];
#pragma unroll
  for (int t = 0; t < 6; ++t) acc[t] = (v8f){};

#pragma unroll 2
  for (int k0 = 0; k0 < 2 * PRED; k0 += 4) {
    const int k  = k0 + koff;                // region never splits inside a step
    const int p2 = (k0 >> 1) + hw;
    const float* src = (k < PRED) ? (arow + k) : (srow + (k - PRED));
    const v2f a = *(const v2f*)src;          // 8B-aligned: k even, rows even
#pragma unroll
    for (int t = 0; t < 6; ++t) {
      const v2f bm = wpp[(size_t)p2 * PRED + t * 16];
      acc[t] = __builtin_amdgcn_wmma_f32_16x16x4_f32(false, a, false, bm,
                                                     (short)0, acc[t],
                                                     false, false);
    }
  }

  const int rbase = mt * 16 + (hw << 3);
#pragma unroll
  for (int t = 0; t < 6; ++t) {
    const int p = t * 16 + l16;
    const float bias = bpred[p];
#pragma unroll
    for (int r = 0; r < 8; ++r) {
      const int rw = rbase + r;
      const int bb = rw >> 6, cc = rw & 63;
      const float xoff = x_enc[(size_t)bb * SEQ * CCH + 511 * CCH + cc];
      out[(size_t)bb * PRED * CCH + (size_t)p * CCH + cc] = acc[t][r] + bias + xoff;
    }
  }
}

// ---------------------------------------------------------------------------
extern "C" void kernel_launch(void* const* d_in, const int* in_sizes, int n_in,
                              void* d_out, int out_size, void* d_ws, size_t ws_size,
                              hipStream_t stream) {
  const float* x_enc = (const float*)d_in[0];
  const int*   index = (const int*)d_in[1];
  const float* Wq    = (const float*)d_in[2];
  const float* bq    = (const float*)d_in[3];
  const float* Wx    = (const float*)d_in[4];
  const float* bx    = (const float*)d_in[5];
  const float* Wprob = (const float*)d_in[6];
  const float* bprob = (const float*)d_in[7];
  const float* Wr0   = (const float*)d_in[8];
  const float* br0   = (const float*)d_in[9];
  const float* Wr1   = (const float*)d_in[10];
  const float* br1   = (const float*)d_in[11];
  const float* Wr2   = (const float*)d_in[12];
  const float* br2   = (const float*)d_in[13];
  const float* Wpred = (const float*)d_in[14];
  const float* bpred = (const float*)d_in[15];
  const float* table = (const float*)d_in[16];
  const float* ang   = (const float*)d_in[17];
  const int*   wires = (const int*)d_in[18];
  const float* gum   = (const float*)d_in[19];
  float* ws  = (float*)d_ws;
  float* out = (float*)d_out;

  prep_kernel<<<(PREP_TOTAL + 255) / 256, 256, 0, stream>>>(
      Wq, Wx, Wr0, br0, Wr1, br1, Wr2, br2, Wpred, ws);
  gemm_base_kernel<<<BATCH, 128, 0, stream>>>(x_enc, bx, bq, ws);
  quantum_kernel<<<BATCH, 64, 0, stream>>>(ws + WS_Q, ang, wires, Wprob, bprob,
                                           gum, (int*)(ws + WS_MSEL));
  gemm_sel_kernel<<<BATCH, 128, 0, stream>>>(table, index, ws);
  gemm_final_kernel<<<BATCH, 128, 0, stream>>>(x_enc, bpred, ws, out);
}